// BilinearInteraction_24747601560016
// MI455X (gfx1250) — compile-verified
//
#include <hip/hip_runtime.h>

// Bilinear interaction ('all' type) for MI455X / gfx1250.
//   x: [B, F=40, D=128] fp32, W: [D, D] fp32
//   vid = x @ W ;  out[b, p, :] = x[b, ii[p], :] * vid[b, jj[p], :],  P = 780
// Output = 818 MB fp32 -> kernel is store-bandwidth bound (~35us @ 23.3 TB/s).
// GEMM (2.7 GFLOP total) done in-LDS with V_WMMA_F32_16X16X4_F32 (fp32 WMMA).

#define NF 40
#define ND 128
#define NP 780           // 40*39/2
#define XS_STRIDE 132    // pad 128->132 floats: A-frag column reads hit distinct banks

typedef float v2f __attribute__((ext_vector_type(2)));
typedef float v4f __attribute__((ext_vector_type(4)));
typedef float v8f __attribute__((ext_vector_type(8)));

__global__ __launch_bounds__(256, 1) void bilinear_wmma_kernel(
    const float* __restrict__ x, const float* __restrict__ W,
    float* __restrict__ out)
{
    __shared__ float xs[48 * XS_STRIDE];       // x[b] padded to 48 rows, stride 132
    __shared__ float Ws[ND * ND];              // W, row major
    __shared__ float vs[48 * ND];              // vid[b] (rows 40..47 are scratch)
    __shared__ unsigned short pij[NP];         // packed (i<<8)|j pair table

    const int tid = threadIdx.x;
    const int b   = blockIdx.x;
    const float* xb = x + (size_t)b * (NF * ND);

    // ---- Stage x[b] into padded LDS (row stride stays 16B-aligned: 132%4==0) ----
    for (int v = tid; v < (NF * ND) / 4; v += 256) {
        int idx = v << 2;
        int r = idx >> 7;          // /128
        int c = idx & 127;
        *(v4f*)&xs[r * XS_STRIDE + c] = *(const v4f*)&xb[idx];
    }
    // Zero pad rows 40..47 (K range read by WMMA is 0..127 only)
    for (int v = tid; v < (8 * ND) / 4; v += 256) {
        int idx = v << 2;
        int r = 40 + (idx >> 7);
        int c = idx & 127;
        *(v4f*)&xs[r * XS_STRIDE + c] = (v4f){0.f, 0.f, 0.f, 0.f};
    }
    // ---- Stage W (64KB, L2-resident across all 2048 blocks) ----
    for (int v = tid; v < (ND * ND) / 4; v += 256)
        ((v4f*)Ws)[v] = ((const v4f*)W)[v];

    // ---- Build triu pair table: row i starts at s(i) = i*(79-i)/2 ----
    for (int p = tid; p < NP; p += 256) {
        int i = 0;
        while (((i + 1) * (79 - (i + 1))) / 2 <= p) ++i;
        int j = p - (i * (79 - i)) / 2 + i + 1;
        pij[p] = (unsigned short)((i << 8) | j);
    }
    __syncthreads();

    // ---- GEMM: vs[0:48, 0:128] = xs[0:48, 0:128] @ Ws via V_WMMA_F32_16X16X4_F32 ----
    // 24 tiles (3 M-tiles x 8 N-tiles); wave w owns N-tile w, all 3 M-tiles.
    // A-frag (16x4 f32, ISA layout): lane-half -> K+{0,2}; the 2 VGPRs -> K+0/K+1.
    // B-frag symmetric: lane(0..15)=N column, lane-half -> K+{0,2}.
    const int wave = tid >> 5;
    const int lane = tid & 31;
    const int lm   = lane & 15;
    const int koff = (lane >> 4) << 1;         // 0 or 2
    const int ncol = wave * 16 + lm;

    v8f acc0 = {}; v8f acc1 = {}; v8f acc2 = {};
    for (int k = 0; k < ND; k += 4) {
        const int kb = k + koff;
        v2f bf;
        bf.x = Ws[kb * ND + ncol];
        bf.y = Ws[(kb + 1) * ND + ncol];
        v2f a0 = *(const v2f*)&xs[( 0 + lm) * XS_STRIDE + kb];
        v2f a1 = *(const v2f*)&xs[(16 + lm) * XS_STRIDE + kb];
        v2f a2 = *(const v2f*)&xs[(32 + lm) * XS_STRIDE + kb];
        acc0 = __builtin_amdgcn_wmma_f32_16x16x4_f32(false, a0, false, bf,
                                                     (short)0, acc0, false, false);
        acc1 = __builtin_amdgcn_wmma_f32_16x16x4_f32(false, a1, false, bf,
                                                     (short)0, acc1, false, false);
        acc2 = __builtin_amdgcn_wmma_f32_16x16x4_f32(false, a2, false, bf,
                                                     (short)0, acc2, false, false);
    }
    // C/D layout: VGPR r, lanes 0-15 -> M=r, lanes 16-31 -> M=r+8, N=lane&15.
    {
        const int half8 = (lane >> 4) << 3;    // 0 or 8
        #pragma unroll
        for (int r = 0; r < 8; ++r) {
            vs[( 0 + r + half8) * ND + ncol] = acc0[r];
            vs[(16 + r + half8) * ND + ncol] = acc1[r];
            vs[(32 + r + half8) * ND + ncol] = acc2[r];
        }
    }
    __syncthreads();

    // ---- Stream output: each wave writes one contiguous 512B pair-row chunk ----
    float* outb = out + (size_t)b * NP * ND;
    const int q    = (tid & 31) << 2;          // float offset within row (v4f lanes)
    const int psub = tid >> 5;                 // 0..7: pair handled by this wave
    for (int base = 0; base < NP; base += 8) {
        const int p = base + psub;
        if (p < NP) {                          // tail: 780 = 97*8 + 4
            const unsigned pr = pij[p];
            const int i = pr >> 8, j = pr & 255;
            v4f xa = *(const v4f*)&xs[i * XS_STRIDE + q];
            v4f vb = *(const v4f*)&vs[j * ND + q];
            v4f r  = xa * vb;
            __builtin_nontemporal_store(r, (v4f*)&outb[(size_t)p * ND + q]);
        }
    }
}

extern "C" void kernel_launch(void* const* d_in, const int* in_sizes, int n_in,
                              void* d_out, int out_size, void* d_ws, size_t ws_size,
                              hipStream_t stream) {
    const float* x = (const float*)d_in[0];   // [B, 40, 128] fp32
    const float* W = (const float*)d_in[1];   // [128, 128] fp32
    float* out = (float*)d_out;               // [B, 780, 128] fp32
    const int B = in_sizes[0] / (NF * ND);
    bilinear_wmma_kernel<<<B, 256, 0, stream>>>(x, W, out);
}